// dy_gat_60696477827350
// MI455X (gfx1250) — compile-verified
//
#include <hip/hip_runtime.h>
#include <math.h>

// ---- problem constants (fixed by the reference setup) ----
#define B_    4
#define T_    12
#define N_    10000
#define E_    160000
#define C_    32        // c_in == c_out == 32
#define TOH_  43
#define TD_   8
#define BT_   (B_*T_)
#define EPS_  9e-15f
#define MAXBLK_PER_BT 32

typedef float v2f __attribute__((ext_vector_type(2)));
typedef float v8f __attribute__((ext_vector_type(8)));

// ---------- order-preserving float<->uint key (for atomicMax on floats) ----------
__device__ __forceinline__ unsigned fkey(float f) {
  unsigned u = __float_as_uint(f);
  return (u & 0x80000000u) ? ~u : (u | 0x80000000u);
}
__device__ __forceinline__ float funkey(unsigned u) {
  return (u & 0x80000000u) ? __uint_as_float(u & 0x7FFFFFFFu) : __uint_as_float(~u);
}

// ---------- 1) zero accumulators (harness poisons d_out/d_ws, never re-zeroes) ----------
__global__ void k_init(float* __restrict__ out, float* __restrict__ rowsum,
                       unsigned* __restrict__ mkey) {
  long i = (long)blockIdx.x * blockDim.x + threadIdx.x;
  const long OUTN = (long)BT_ * N_ * C_;
  const long RN   = (long)BT_ * N_;
  if (i < OUTN) out[i] = 0.0f;
  if (i < RN)   rowsum[i] = 0.0f;
  if (i < BT_)  mkey[i] = 0u;   // key 0 == below every real float value
}

// ---------- 2) attn[b, 0..63] = (timeoh[b] @ time_w) @ atten_pool ----------
__global__ void k_attn(const float* __restrict__ timeoh, const float* __restrict__ time_w,
                       const float* __restrict__ atten_pool, float* __restrict__ attn) {
  int b = blockIdx.x;          // 0..3
  int j = threadIdx.x;         // 0..63
  float tmp[TD_];
#pragma unroll
  for (int k = 0; k < TD_; ++k) {
    float s = 0.f;
    for (int i = 0; i < TOH_; ++i) s += timeoh[b*TOH_ + i] * time_w[i*TD_ + k];
    tmp[k] = s;
  }
  float a = 0.f;
#pragma unroll
  for (int k = 0; k < TD_; ++k) a += tmp[k] * atten_pool[k*(2*C_) + j];
  attn[b*(2*C_) + j] = a;
}

// ---------- 3) h = x @ W + bias via V_WMMA_F32_16X16X4_F32 ----------
// One wave per 16-row tile; 8 K-steps (K=4) x 2 column tiles = 16 WMMA/tile.
// Grid is exact (30000 waves), so EXEC is all-ones at every WMMA.
__global__ void k_gemm(const float* __restrict__ x, const float* __restrict__ W,
                       const float* __restrict__ bias, float* __restrict__ h) {
  int wave = (int)((blockIdx.x * blockDim.x + threadIdx.x) >> 5);
  int lane = threadIdx.x & 31;
  int bt   = wave / (N_ / 16);
  int tile = wave % (N_ / 16);
  long rowbase = (long)bt * N_ + (long)tile * 16;
  const float* xa = x + rowbase * C_;

  int halfsel = lane >> 4;       // 0: K+0/K+1 (A) / rows 0..7 (D); 1: K+2/K+3 / rows 8..15
  int lrow    = lane & 15;       // A: row within tile; B/D: column within 16-col tile

  v8f acc0 = {};                 // output cols 0..15
  v8f acc1 = {};                 // output cols 16..31
#pragma unroll
  for (int kk = 0; kk < C_/4; ++kk) {
    int kb = kk*4 + halfsel*2;
    v2f a;  a.x  = xa[lrow*C_ + kb];          a.y  = xa[lrow*C_ + kb + 1];
    v2f b0; b0.x = W[kb*C_ + lrow];           b0.y = W[(kb+1)*C_ + lrow];
    v2f b1; b1.x = W[kb*C_ + 16 + lrow];      b1.y = W[(kb+1)*C_ + 16 + lrow];
    acc0 = __builtin_amdgcn_wmma_f32_16x16x4_f32(false, a, false, b0, (short)0, acc0, false, false);
    acc1 = __builtin_amdgcn_wmma_f32_16x16x4_f32(false, a, false, b1, (short)0, acc1, false, false);
  }
  // D layout: VGPR r -> row r (lanes 0-15, col=lane) / row r+8 (lanes 16-31, col=lane-16)
  float bias0 = bias[lrow];
  float bias1 = bias[16 + lrow];
  float* hp = h + rowbase * C_;
#pragma unroll
  for (int r = 0; r < 8; ++r) {
    int row = r + halfsel*8;
    hp[row*C_ + lrow]      = acc0[r] + bias0;
    hp[row*C_ + 16 + lrow] = acc1[r] + bias1;
  }
}

// ---------- 4) per-node scores: s_src = h . attn[:32], s_dst = h . attn[32:] ----------
__global__ void k_score(const float* __restrict__ h, const float* __restrict__ attn,
                        float* __restrict__ s_src, float* __restrict__ s_dst) {
  long i = (long)blockIdx.x * blockDim.x + threadIdx.x;    // node index in [0, BT*N)
  if (i >= (long)BT_ * N_) return;
  int b = (int)(i / ((long)T_ * N_));
  const float* hv = h + i * C_;
  const float* as = attn + b * (2*C_);
  float ss = 0.f, sd = 0.f;
#pragma unroll
  for (int c = 0; c < C_; ++c) {
    float v = hv[c];
    ss += v * as[c];
    sd += v * as[C_ + c];
  }
  s_src[i] = ss;
  s_dst[i] = sd;
}

// ---------- 5) per-(b,t) max over edges of leaky_relu(s_src[src] + s_dst[dst]) ----------
__global__ void k_max(const float* __restrict__ s_src, const float* __restrict__ s_dst,
                      const int* __restrict__ edge, unsigned* __restrict__ mkey) {
  int bt  = blockIdx.x / MAXBLK_PER_BT;
  int sub = blockIdx.x % MAXBLK_PER_BT;
  const float* ss = s_src + (long)bt * N_;
  const float* sd = s_dst + (long)bt * N_;
  const int* src = edge;         // edge[0,:]
  const int* dst = edge + E_;    // edge[1,:]
  float m = -3.4e38f;
  for (int e = sub * blockDim.x + threadIdx.x; e < E_; e += MAXBLK_PER_BT * blockDim.x) {
    float v = ss[src[e]] + sd[dst[e]];
    v = v > 0.f ? v : 0.01f * v;           // leaky_relu(., 0.01)
    m = fmaxf(m, v);
  }
  __shared__ float red[8];
#pragma unroll
  for (int off = 16; off > 0; off >>= 1) m = fmaxf(m, __shfl_down(m, off, 32));
  int lane = threadIdx.x & 31, wid = threadIdx.x >> 5;
  if (lane == 0) red[wid] = m;
  __syncthreads();
  if (threadIdx.x == 0) {
    float mm = red[0];
    for (int w = 1; w < (int)(blockDim.x >> 5); ++w) mm = fmaxf(mm, red[w]);
    atomicMax(&mkey[bt], fkey(mm));        // native GLOBAL_ATOMIC_MAX_U32
  }
}

// ---------- 6) edge scatter: one wave32 per (b,t,e); lane = channel ----------
__global__ void k_scatter(const float* __restrict__ h, const float* __restrict__ s_src,
                          const float* __restrict__ s_dst, const int* __restrict__ edge,
                          const unsigned* __restrict__ mkey,
                          float* __restrict__ out, float* __restrict__ rowsum) {
  long gw  = ((long)blockIdx.x * blockDim.x + threadIdx.x) >> 5;  // (bt, e) pair
  int lane = threadIdx.x & 31;                                     // channel
  int bt = (int)(gw / E_);
  int e  = (int)(gw % E_);
  int s  = edge[e];
  int d  = edge[E_ + e];
  long base = (long)bt * N_;
  float v = s_src[base + s] + s_dst[base + d];
  v = v > 0.f ? v : 0.01f * v;
  float w = expf(v - funkey(mkey[bt]));
  float hv = h[(base + d) * C_ + lane];                 // coalesced 128B gather
  atomicAdd(&out[(base + s) * C_ + lane], w * hv);      // GLOBAL_ATOMIC_ADD_F32, L2-resident
  if (lane == 0) atomicAdd(&rowsum[base + s], w);
}

// ---------- 7) normalize: out /= (rowsum + EPS) ----------
__global__ void k_norm(float* __restrict__ out, const float* __restrict__ rowsum) {
  long i = (long)blockIdx.x * blockDim.x + threadIdx.x;
  if (i >= (long)BT_ * N_ * C_) return;
  out[i] = out[i] / (rowsum[i >> 5] + EPS_);            // C_ == 32
}

extern "C" void kernel_launch(void* const* d_in, const int* in_sizes, int n_in,
                              void* d_out, int out_size, void* d_ws, size_t ws_size,
                              hipStream_t stream) {
  const float* x          = (const float*)d_in[0];
  const float* timeoh     = (const float*)d_in[1];
  const int*   edge       = (const int*)  d_in[2];
  const float* mlp_w      = (const float*)d_in[3];
  const float* mlp_b      = (const float*)d_in[4];
  const float* time_w     = (const float*)d_in[5];
  const float* atten_pool = (const float*)d_in[6];
  float* out = (float*)d_out;

  // workspace layout (floats): h | s_src | s_dst | rowsum | attn | mkey
  float* ws     = (float*)d_ws;
  float* h      = ws;                                 // BT*N*C   = 15,360,000
  float* s_src  = h      + (size_t)BT_ * N_ * C_;     // BT*N     =    480,000
  float* s_dst  = s_src  + (size_t)BT_ * N_;
  float* rowsum = s_dst  + (size_t)BT_ * N_;
  float* attn   = rowsum + (size_t)BT_ * N_;          // B*64     = 256
  unsigned* mkey = (unsigned*)(attn + B_ * 2 * C_);   // BT       = 48

  const long OUTN = (long)BT_ * N_ * C_;              // 15,360,000 (divisible by 256)

  k_init <<< (unsigned)((OUTN + 255) / 256), 256, 0, stream >>>(out, rowsum, mkey);
  k_attn <<< B_, 2 * C_, 0, stream >>>(timeoh, time_w, atten_pool, attn);

  const int waves = BT_ * (N_ / 16);                  // 30000 tiles, exact
  k_gemm <<< (unsigned)(waves / 8), 256, 0, stream >>>(x, mlp_w, mlp_b, h);

  k_score<<< (unsigned)(((long)BT_ * N_ + 255) / 256), 256, 0, stream >>>(h, attn, s_src, s_dst);
  k_max  <<< BT_ * MAXBLK_PER_BT, 256, 0, stream >>>(s_src, s_dst, edge, mkey);

  const long totw = (long)BT_ * E_ * 32;              // 245,760,000 (exact multiple of 256)
  k_scatter<<< (unsigned)(totw / 256), 256, 0, stream >>>(h, s_src, s_dst, edge, mkey, out, rowsum);

  k_norm <<< (unsigned)((OUTN + 255) / 256), 256, 0, stream >>>(out, rowsum);
}